// CoarsePyramid_54528904790790
// MI455X (gfx1250) — compile-verified
//
#include <hip/hip_runtime.h>
#include <hip/hip_bf16.h>

// ---------------------------------------------------------------------------
// CoarsePyramid head for MI455X (gfx1250), fp32 via V_WMMA_F32_16X16X4_F32.
// Shapes: B=8, C=512, T=64, TF=256, CCONF=400, GROUPS=32, CAT=2448.
// ---------------------------------------------------------------------------

#define BATCH  8
#define CH     512
#define TT     64
#define TFRAME 256
#define CCONF  400
#define NGROUP 32
#define CATCH  2448      // 512 + 1024 + 512 + 400
#define EPSV   1e-5f

typedef __attribute__((ext_vector_type(2))) float v2f;
typedef __attribute__((ext_vector_type(8))) float v8f;

// ---------------------------------------------------------------------------
// GEMM + bias:  Y[b, m0:m0+16, 0:64] = W[M,K] @ X[b,K,64] + bias
// One wave computes a 16x64 strip: 4 accumulators (n0 = 0,16,32,48) share each
// A-tile load, K marched in steps of 4 with v_wmma_f32_16x16x4_f32.
// All addressing is uniform-base + 32-bit element offset (GVS saddr form).
// ---------------------------------------------------------------------------
__global__ void gemm_bias_wmma_f32(const float* __restrict__ W,   // [M,K] row-major
                                   const float* __restrict__ X,   // [B,K,64]
                                   const float* __restrict__ bias,// [M]
                                   float* __restrict__ Y,         // per-batch base + y_bstride
                                   int M, int K,
                                   int x_bstride, int y_bstride,
                                   int total_waves)
{
    const int lane  = threadIdx.x & 31;
    const int wave  = threadIdx.x >> 5;
    const int gwave = blockIdx.x * (blockDim.x >> 5) + wave;
    if (gwave >= total_waves) return;            // uniform per wave: EXEC stays all-1s

    const int mt = M >> 4;
    const int b  = gwave / mt;
    const int m0 = (gwave - b * mt) << 4;

    const int lm = lane & 15;                    // N col / M row within half
    const int lh = lane >> 4;                    // lane half selector

    // 32-bit element offsets from uniform bases (saddr-form loads)
    const int woff = (m0 + lm) * K + 2 * lh;               // A: W[m0+lm, k + 2*lh]
    const int xoff = b * x_bstride + 2 * lh * TT + lm;     // B: X[b, k + 2*lh + g, n0+lm]

    // accumulators = bias (D vgpr g holds row m0+g+8*lh; bias independent of N)
    v8f acc0, acc1, acc2, acc3;
#pragma unroll
    for (int g = 0; g < 8; ++g) {
        float bv = bias[m0 + g + 8 * lh];
        acc0[g] = bv; acc1[g] = bv; acc2[g] = bv; acc3[g] = bv;
    }

#pragma unroll 2
    for (int k = 0; k < K; k += 4) {
        // A: 16x4 fp32 tile of W (one contiguous float2 per lane), shared by 4 WMMAs
        v2f a = *(const v2f*)(W + (woff + k));
        __builtin_prefetch(W + (woff + k) + 128, 0, 1);   // global_prefetch_b8

        const int xk = xoff + k * TT;
        v2f b0; b0.x = X[xk +  0]; b0.y = X[xk + TT +  0];
        v2f b1; b1.x = X[xk + 16]; b1.y = X[xk + TT + 16];
        v2f b2; b2.x = X[xk + 32]; b2.y = X[xk + TT + 32];
        v2f b3; b3.x = X[xk + 48]; b3.y = X[xk + TT + 48];

        acc0 = __builtin_amdgcn_wmma_f32_16x16x4_f32(false, a, false, b0, (short)0, acc0, false, false);
        acc1 = __builtin_amdgcn_wmma_f32_16x16x4_f32(false, a, false, b1, (short)0, acc1, false, false);
        acc2 = __builtin_amdgcn_wmma_f32_16x16x4_f32(false, a, false, b2, (short)0, acc2, false, false);
        acc3 = __builtin_amdgcn_wmma_f32_16x16x4_f32(false, a, false, b3, (short)0, acc3, false, false);
    }

    const int ybase = b * y_bstride + lm;
#pragma unroll
    for (int g = 0; g < 8; ++g) {
        const int row = (m0 + g + 8 * lh) * TT;
        Y[ybase + row +  0] = acc0[g];
        Y[ybase + row + 16] = acc1[g];
        Y[ybase + row + 32] = acc2[g];
        Y[ybase + row + 48] = acc3[g];
    }
}

// ---------------------------------------------------------------------------
// In-place GroupNorm + ReLU.  One block per (batch, group); a group's channel
// slab is contiguous within a batch (channel stride == N), so reduce linearly.
// ---------------------------------------------------------------------------
__global__ void groupnorm_relu(float* __restrict__ Y,
                               const float* __restrict__ gamma,
                               const float* __restrict__ beta,
                               int C, int N, int bstride)
{
    const int b   = blockIdx.x / NGROUP;
    const int g   = blockIdx.x - b * NGROUP;
    const int cpg = C / NGROUP;
    const int n   = cpg * N;

    float* __restrict__ base = Y + (size_t)b * bstride + (size_t)g * cpg * N;

    float s = 0.f, ss = 0.f;
    for (int i = threadIdx.x; i < n; i += blockDim.x) {
        float v = base[i];
        s += v; ss += v * v;
    }

    __shared__ float r0[256];
    __shared__ float r1[256];
    r0[threadIdx.x] = s;
    r1[threadIdx.x] = ss;
    __syncthreads();
    for (int off = 128; off > 0; off >>= 1) {
        if ((int)threadIdx.x < off) {
            r0[threadIdx.x] += r0[threadIdx.x + off];
            r1[threadIdx.x] += r1[threadIdx.x + off];
        }
        __syncthreads();
    }
    const float inv_n = 1.0f / (float)n;
    const float mu    = r0[0] * inv_n;
    const float var   = r1[0] * inv_n - mu * mu;
    const float rstd  = rsqrtf(var + EPSV);

    for (int i = threadIdx.x; i < n; i += blockDim.x) {
        int c  = g * cpg + i / N;
        float v = (base[i] - mu) * rstd * gamma[c] + beta[c];
        base[i] = fmaxf(v, 0.f);
    }
}

// ---------------------------------------------------------------------------
// Boundary max pooling.  feature [B,CC,Tin], seg [B,NP,4] -> out [B,CC,NP].
// First CC/2 channels pool over [s0,s1], rest over [s2,s3]; empty -> 0.
// Output written with batch stride o_bstride (channel stride = NP).
// ---------------------------------------------------------------------------
__global__ void boundary_pool(const float* __restrict__ F,
                              const float* __restrict__ seg,
                              float* __restrict__ O,
                              int CC, int Tin, int NP, int o_bstride, int total)
{
    int idx = blockIdx.x * blockDim.x + threadIdx.x;
    if (idx >= total) return;

    const int n = idx % NP;
    const int c = (idx / NP) % CC;
    const int b = idx / (NP * CC);

    const float* s = seg + ((size_t)b * NP + n) * 4;
    const int k  = (c < (CC >> 1)) ? 0 : 2;
    int lo = (int)floorf(s[k]);
    int hi = (int)floorf(s[k + 1]);
    lo = min(max(lo, 0), Tin - 1);
    hi = min(max(hi, 0), Tin - 1);

    const float* row = F + ((size_t)b * CC + c) * Tin;
    float m = -__builtin_inff();
    for (int t = lo; t <= hi; ++t) m = fmaxf(m, row[t]);
    if (hi < lo) m = 0.f;

    O[(size_t)b * o_bstride + (size_t)c * NP + n] = m;
}

// ---------------------------------------------------------------------------
// Copy conf_result_feature [B,400,64] into its slice of the cat buffer.
// ---------------------------------------------------------------------------
__global__ void copy_conf(const float* __restrict__ src, float* __restrict__ dst,
                          int total)
{
    int idx = blockIdx.x * blockDim.x + threadIdx.x;
    if (idx >= total) return;
    const int t = idx % TT;
    const int c = (idx / TT) % CCONF;
    const int b = idx / (TT * CCONF);
    dst[(size_t)b * (CATCH * TT) + (size_t)c * TT + t] =
        src[(size_t)b * (CCONF * TT) + (size_t)c * TT + t];
}

// ---------------------------------------------------------------------------
extern "C" void kernel_launch(void* const* d_in, const int* in_sizes, int n_in,
                              void* d_out, int out_size, void* d_ws, size_t ws_size,
                              hipStream_t stream) {
    const float* feature   = (const float*)d_in[0];   // [8,512,64]
    const float* frame_f   = (const float*)d_in[1];   // [8,512,256]
    const float* segments  = (const float*)d_in[2];   // [8,64,4]
    const float* frame_seg = (const float*)d_in[3];   // [8,64,4]
    const float* conf_f    = (const float*)d_in[4];   // [8,400,64]
    const float* w_cur  = (const float*)d_in[5];
    const float* b_cur  = (const float*)d_in[6];
    const float* g_cur  = (const float*)d_in[7];
    const float* be_cur = (const float*)d_in[8];
    const float* w_lr   = (const float*)d_in[9];
    const float* b_lr   = (const float*)d_in[10];
    const float* g_lr   = (const float*)d_in[11];
    const float* be_lr  = (const float*)d_in[12];
    const float* w_roi  = (const float*)d_in[13];
    const float* b_roi  = (const float*)d_in[14];
    const float* g_roi  = (const float*)d_in[15];
    const float* be_roi = (const float*)d_in[16];
    const float* w_prop = (const float*)d_in[17];
    const float* b_prop = (const float*)d_in[18];
    const float* g_prop = (const float*)d_in[19];
    const float* be_prop= (const float*)d_in[20];

    float* out   = (float*)d_out;                     // [8,512,64]
    float* feat2 = out + BATCH * CH * TT;             // [8,1024,64] (2nd output)

    // workspace: cat [8,2448,64] then prop_roi_pre [8,512,64]
    float* cat      = (float*)d_ws;
    float* pool_pre = cat + (size_t)BATCH * CATCH * TT;

    float* cat_roi   = cat;                            // channels    0..511
    float* cat_propf = cat + 512 * TT;                 // channels  512..1535
    float* cat_short = cat + 1536 * TT;                // channels 1536..2047
    float* cat_conf  = cat + 2048 * TT;                // channels 2048..2447

    const int catstride = CATCH * TT;
    const dim3 blk128(128), blk256(256);
    const int wpb = 4;                                 // waves per block

    // 1) fm_short = CGR(feature, w_cur) -> cat channels 1536..2047
    {
        int waves = BATCH * (CH / 16);                 // 256
        gemm_bias_wmma_f32<<<(waves + wpb - 1) / wpb, blk128, 0, stream>>>(
            w_cur, feature, b_cur, cat_short, CH, CH, CH * TT, catstride, waves);
        groupnorm_relu<<<BATCH * NGROUP, blk256, 0, stream>>>(cat_short, g_cur, be_cur,
                                                              CH, TT, catstride);
    }

    // 2) feat2 = CGR(feature, w_lr) -> second output buffer
    {
        int M = 2 * CH;
        int waves = BATCH * (M / 16);                  // 512
        gemm_bias_wmma_f32<<<(waves + wpb - 1) / wpb, blk128, 0, stream>>>(
            w_lr, feature, b_lr, feat2, M, CH, CH * TT, M * TT, waves);
        groupnorm_relu<<<BATCH * NGROUP, blk256, 0, stream>>>(feat2, g_lr, be_lr,
                                                              M, TT, M * TT);
    }

    // 3) prop_feature = pool(feat2, segments) -> cat channels 512..1535
    {
        int total = BATCH * (2 * CH) * TT;             // 524288
        boundary_pool<<<(total + 255) / 256, blk256, 0, stream>>>(
            feat2, segments, cat_propf, 2 * CH, TT, TT, catstride, total);
    }

    // 4) prop_roi_pre = pool(frame_level_feature, frame_segments)
    {
        int total = BATCH * CH * TT;                   // 262144
        boundary_pool<<<(total + 255) / 256, blk256, 0, stream>>>(
            frame_f, frame_seg, pool_pre, CH, TFRAME, TT, CH * TT, total);
    }

    // 5) prop_roi = CGR(prop_roi_pre, w_roi) -> cat channels 0..511
    {
        int waves = BATCH * (CH / 16);
        gemm_bias_wmma_f32<<<(waves + wpb - 1) / wpb, blk128, 0, stream>>>(
            w_roi, pool_pre, b_roi, cat_roi, CH, CH, CH * TT, catstride, waves);
        groupnorm_relu<<<BATCH * NGROUP, blk256, 0, stream>>>(cat_roi, g_roi, be_roi,
                                                              CH, TT, catstride);
    }

    // 6) conf slice -> cat channels 2048..2447
    {
        int total = BATCH * CCONF * TT;                // 204800
        copy_conf<<<(total + 255) / 256, blk256, 0, stream>>>(conf_f, cat_conf, total);
    }

    // 7) out = CGR(cat, w_prop)  (K = 2448)
    {
        int waves = BATCH * (CH / 16);
        gemm_bias_wmma_f32<<<(waves + wpb - 1) / wpb, blk128, 0, stream>>>(
            w_prop, cat, b_prop, out, CH, CATCH, catstride, CH * TT, waves);
        groupnorm_relu<<<BATCH * NGROUP, blk256, 0, stream>>>(out, g_prop, be_prop,
                                                              CH, TT, CH * TT);
    }
}